// AdditiveLowRankRoute_29669634081462
// MI455X (gfx1250) — compile-verified
//
#include <hip/hip_runtime.h>
#include <hip/hip_bf16.h>

// ---------------------------------------------------------------------------
// AdditiveLowRankRoute for MI455X (gfx1250, wave32)
//
//   out[b,s,t] = sum_w silu(ps[b,s,w]*pt[b,t,w]) * w_int[w]
//              + s_lin[b,s] + t_lin[b,t] + bias
//
// Pipeline:
//   1) proj_wmma:   ps_half = 0.5*(src @ Ws^T), pt = tgt @ Wt^T   (V_WMMA_F32_16X16X4_F32)
//   2) rowdot:      s_lin = 2*dot(ps_half,ws_out), t_lin = dot(pt,wt_out)
//   3) route_main:  tiled nonlinear bilinear interaction, silu via v_tanh_f32:
//                   with h = 0.5*p:  silu(p) = h + h*tanh(h)
// ---------------------------------------------------------------------------

#define B_DIM 2
#define S_DIM 4096
#define T_DIM 4096
#define SRC_K 512
#define W_DIM 128

typedef __attribute__((ext_vector_type(2))) float v2f;
typedef __attribute__((ext_vector_type(8))) float v8f;

// silu(2h) = h * (1 + tanh(h)), h = 0.5 * p
__device__ __forceinline__ float silu_from_half(float h) {
#if __has_builtin(__builtin_amdgcn_tanhf)
  float t = __builtin_amdgcn_tanhf(h);          // v_tanh_f32 (TRANS)
  return __builtin_fmaf(h, t, h);
#else
  // silu(2h) = 2h * sigmoid(2h) = 2h / (1 + exp2(-2h*log2 e))
  float e = __builtin_amdgcn_exp2f(h * -2.885390081777927f);
  return (h + h) * __builtin_amdgcn_rcpf(1.0f + e);
#endif
}

// ---------------------------------------------------------------------------
// Kernel 1: P[M,N] = scale * (X[M,K] @ W[N,K]^T) via V_WMMA_F32_16X16X4_F32.
// One wave32 owns one 16x16 C tile; K stepped by 4.
// ISA VGPR layouts (CDNA5 ISA):
//   A 16x4 : lane<16 -> M=lane, {K=k,k+1}; lane>=16 -> M=lane-16, {K=k+2,k+3}
//   B 4x16 : lane<16 -> N=lane, {K=k,k+1}; lane>=16 -> N=lane-16, {K=k+2,k+3}
//   C 16x16: VGPR i, lane<16 -> (M=i, N=lane); lane>=16 -> (M=i+8, N=lane-16)
// ---------------------------------------------------------------------------
__global__ __launch_bounds__(256) void proj_wmma(const float* __restrict__ X,
                                                 const float* __restrict__ W,
                                                 float* __restrict__ P,
                                                 float scale, int M, int N, int K) {
  const int wave = threadIdx.x >> 5;
  const int lane = threadIdx.x & 31;
  const int gw   = blockIdx.x * 8 + wave;       // global wave (tile) index
  const int ntn  = N >> 4;
  const int m0   = (gw / ntn) << 4;
  const int n0   = (gw % ntn) << 4;
  const int half = lane >> 4;                   // 0: K pair {0,1}; 1: {2,3}
  const int r    = lane & 15;

  const float* aptr = X + (size_t)(m0 + r) * K + 2 * half;
  const float* bptr = W + (size_t)(n0 + r) * K + 2 * half;

  v8f c = {};
  for (int k = 0; k < K; k += 4) {
    v2f av = *(const v2f*)(aptr + k);
    v2f bv = *(const v2f*)(bptr + k);
    // 8 args: (neg_a, A, neg_b, B, c_mod, C, reuse_a, reuse_b)
    c = __builtin_amdgcn_wmma_f32_16x16x4_f32(false, av, false, bv,
                                              (short)0, c, false, false);
  }

#pragma unroll
  for (int i = 0; i < 8; ++i) {
    P[(size_t)(m0 + half * 8 + i) * N + n0 + r] = c[i] * scale;
  }
}

// ---------------------------------------------------------------------------
// Kernel 2: o[m] = scale * dot(P[m, 0:128], w[0:128])
// ---------------------------------------------------------------------------
__global__ __launch_bounds__(256) void rowdot(const float* __restrict__ P,
                                              const float* __restrict__ wv,
                                              float* __restrict__ o,
                                              float scale, int M) {
  int m = blockIdx.x * 256 + threadIdx.x;
  if (m >= M) return;
  const float4* row = (const float4*)(P + (size_t)m * W_DIM);
  const float4* wp  = (const float4*)wv;
  float acc = 0.f;
#pragma unroll
  for (int i = 0; i < W_DIM / 4; ++i) {
    float4 p = row[i];
    float4 w = wp[i];
    acc = __builtin_fmaf(p.x, w.x, acc);
    acc = __builtin_fmaf(p.y, w.y, acc);
    acc = __builtin_fmaf(p.z, w.z, acc);
    acc = __builtin_fmaf(p.w, w.w, acc);
  }
  o[m] = acc * scale;
}

// ---------------------------------------------------------------------------
// Kernel 3: main interaction. 64(s) x 64(t) tile per 256-thread block,
// 4x4 micro-tile per thread, full W=128 reduction. LDS = 64 KB exactly.
// Compute-bound on v_tanh_f32 (1 tanh + 3 VALU per term), memory is noise.
// ---------------------------------------------------------------------------
#define TS 64
#define TT 64

__global__ __launch_bounds__(256) void route_main(const float* __restrict__ psh,
                                                  const float* __restrict__ pt,
                                                  const float* __restrict__ s_lin,
                                                  const float* __restrict__ t_lin,
                                                  const float* __restrict__ w_int,
                                                  const float* __restrict__ bias,
                                                  float* __restrict__ out) {
  __shared__ float sP[TS][W_DIM];   // 32 KB
  __shared__ float sT[TT][W_DIM];   // 32 KB

  const int b  = blockIdx.z;
  const int s0 = blockIdx.y * TS;
  const int t0 = blockIdx.x * TT;
  const int tid = threadIdx.x;

  // Stage tiles: float4 per thread, fully coalesced rows.
  const float* gP = psh + ((size_t)b * S_DIM + s0) * W_DIM;
  const float* gT = pt  + ((size_t)b * T_DIM + t0) * W_DIM;
  for (int idx = tid; idx < TS * (W_DIM / 4); idx += 256) {
    int row = idx >> 5;           // W_DIM/4 == 32
    int w4  = idx & 31;
    ((float4*)&sP[row][0])[w4] = ((const float4*)(gP + (size_t)row * W_DIM))[w4];
    ((float4*)&sT[row][0])[w4] = ((const float4*)(gT + (size_t)row * W_DIM))[w4];
  }
  __syncthreads();

  const int tx = tid & 15;        // 4 t-columns
  const int ty = tid >> 4;        // 4 s-rows

  float acc[4][4] = {};

  for (int w = 0; w < W_DIM; w += 4) {
    float4 wv = *(const float4*)(w_int + w);   // uniform -> scalar-cached
    float4 a[4], bt[4];
#pragma unroll
    for (int i = 0; i < 4; ++i) a[i]  = *(const float4*)&sP[ty * 4 + i][w];
#pragma unroll
    for (int j = 0; j < 4; ++j) bt[j] = *(const float4*)&sT[tx * 4 + j][w];

#pragma unroll
    for (int i = 0; i < 4; ++i) {
#pragma unroll
      for (int j = 0; j < 4; ++j) {
        acc[i][j] = __builtin_fmaf(silu_from_half(a[i].x * bt[j].x), wv.x, acc[i][j]);
        acc[i][j] = __builtin_fmaf(silu_from_half(a[i].y * bt[j].y), wv.y, acc[i][j]);
        acc[i][j] = __builtin_fmaf(silu_from_half(a[i].z * bt[j].z), wv.z, acc[i][j]);
        acc[i][j] = __builtin_fmaf(silu_from_half(a[i].w * bt[j].w), wv.w, acc[i][j]);
      }
    }
  }

  const float bi = bias[0];
  float sl[4], tl[4];
#pragma unroll
  for (int i = 0; i < 4; ++i) sl[i] = s_lin[b * S_DIM + s0 + ty * 4 + i] + bi;
#pragma unroll
  for (int j = 0; j < 4; ++j) tl[j] = t_lin[b * T_DIM + t0 + tx * 4 + j];

#pragma unroll
  for (int i = 0; i < 4; ++i) {
    float4 v;
    v.x = acc[i][0] + sl[i] + tl[0];
    v.y = acc[i][1] + sl[i] + tl[1];
    v.z = acc[i][2] + sl[i] + tl[2];
    v.w = acc[i][3] + sl[i] + tl[3];
    float* op = out + ((size_t)b * S_DIM + s0 + ty * 4 + i) * T_DIM + t0 + tx * 4;
    *(float4*)op = v;
  }
}

// ---------------------------------------------------------------------------
extern "C" void kernel_launch(void* const* d_in, const int* in_sizes, int n_in,
                              void* d_out, int out_size, void* d_ws, size_t ws_size,
                              hipStream_t stream) {
  const float* source_val = (const float*)d_in[0];   // [B,S,SRC]
  const float* target_val = (const float*)d_in[1];   // [B,T,DST]
  const float* Ws         = (const float*)d_in[2];   // [W,SRC]
  const float* Wt         = (const float*)d_in[3];   // [W,DST]
  const float* ws_out     = (const float*)d_in[4];   // [W]
  const float* wt_out     = (const float*)d_in[5];   // [W]
  const float* w_int      = (const float*)d_in[6];   // [W]
  const float* bias       = (const float*)d_in[7];   // scalar
  float* out = (float*)d_out;

  const int M = B_DIM * S_DIM;                       // 8192 rows for both projections

  // Workspace layout (needs ~8.1 MB)
  float* psh   = (float*)d_ws;                       // 0.5*ps  [M,128]  4 MB
  float* ptb   = psh + (size_t)M * W_DIM;            // pt      [M,128]  4 MB
  float* slin  = ptb + (size_t)M * W_DIM;            // [M]
  float* tlin  = slin + M;                           // [M]

  // 1) projections via f32 WMMA: (M/16)*(W/16) = 4096 tiles, 8 waves/block
  const int proj_blocks = (M / 16) * (W_DIM / 16) / 8;   // 512
  proj_wmma<<<proj_blocks, 256, 0, stream>>>(source_val, Ws, psh, 0.5f, M, W_DIM, SRC_K);
  proj_wmma<<<proj_blocks, 256, 0, stream>>>(target_val, Wt, ptb, 1.0f, M, W_DIM, SRC_K);

  // 2) linear terms (s_lin needs x2 to undo the 0.5 folded into psh)
  rowdot<<<(M + 255) / 256, 256, 0, stream>>>(psh, ws_out, slin, 2.0f, M);
  rowdot<<<(M + 255) / 256, 256, 0, stream>>>(ptb, wt_out, tlin, 1.0f, M);

  // 3) main interaction
  dim3 grid(T_DIM / TT, S_DIM / TS, B_DIM);
  route_main<<<grid, 256, 0, stream>>>(psh, ptb, slin, tlin, w_int, bias, out);
}